// SGDRHierarchicalMemory_47467978555545
// MI455X (gfx1250) — compile-verified
//
#include <hip/hip_runtime.h>
#include <math.h>
#include <stdint.h>

typedef __attribute__((ext_vector_type(2))) float v2f;
typedef __attribute__((ext_vector_type(8))) float v8f;
typedef int v4i_ __attribute__((vector_size(16)));   // matches async-LDS builtin param

#define D_MODEL 2048
#define D_STATE 128
#define BATCH   8
#define SEQ     1024
#define ROWS    (BATCH * SEQ)   // 8192

#if defined(__has_builtin)
#  if __has_builtin(__builtin_amdgcn_global_load_async_to_lds_b128)
#    define HAVE_ASYNC_LDS 1
#  endif
#endif
#ifndef HAVE_ASYNC_LDS
#  define HAVE_ASYNC_LDS 0
#endif

__device__ __forceinline__ float sigmoidf_(float x) { return 1.0f / (1.0f + expf(-x)); }
__device__ __forceinline__ float gelu_exact(float x) {
    return 0.5f * x * (1.0f + erff(x * 0.70710678118654752f));
}

#if HAVE_ASYNC_LDS
__device__ __forceinline__ void async_copy_b128(const float* g, float* l)
{
    __builtin_amdgcn_global_load_async_to_lds_b128(
        (__attribute__((address_space(1))) v4i_*)(uintptr_t)g,
        (__attribute__((address_space(3))) v4i_*)(uint32_t)(uintptr_t)l,
        0, 0);
}
#endif

// -------------------------------------------------------------------------
// fp32 WMMA GEMM: C[M,N] = A[M,K] @ B[K,N]  (+bias, +gelu)
// One wave per workgroup; each wave register-tiles 2 row-tiles x 4 col-tiles
// (32x64 of C, 8 accumulators) so each A fragment feeds 4 WMMAs and each B
// fragment feeds 2.   mode: 0 = none, 1 = +bias, 2 = +bias then exact GELU
// block = 32 threads; grid = (M/32, N/64)
// -------------------------------------------------------------------------
__global__ __launch_bounds__(32)
void gemm_wmma_f32_kernel(const float* __restrict__ A, const float* __restrict__ B,
                          const float* __restrict__ bias, float* __restrict__ C,
                          int M, int N, int K, int mode)
{
    const int lane = threadIdx.x & 31;
    const int m    = lane & 15;     // row-within-tile (A) / col-within-tile (B,C)
    const int h    = lane >> 4;     // half-wave select
    const int row0 = blockIdx.x * 32;
    const int col0 = blockIdx.y * 64;
    if (row0 >= M || col0 >= N) return;

    const float* Arow0 = A + (size_t)(row0 + m) * K;
    const float* Arow1 = A + (size_t)(row0 + 16 + m) * K;

    v8f acc[2][4];
    #pragma unroll
    for (int rt = 0; rt < 2; ++rt)
        #pragma unroll
        for (int ct = 0; ct < 4; ++ct)
            acc[rt][ct] = (v8f){0.f,0.f,0.f,0.f,0.f,0.f,0.f,0.f};

    for (int k = 0; k < K; k += 4) {
        if ((k & 63) == 0) {
            __builtin_prefetch(Arow0 + k + 64, 0, 1);   // global_prefetch_b8
            __builtin_prefetch(Arow1 + k + 64, 0, 1);
        }
        const int ka = k + 2 * h;
        v2f a0, a1;
        a0.x = Arow0[ka]; a0.y = Arow0[ka + 1];
        a1.x = Arow1[ka]; a1.y = Arow1[ka + 1];

        const float* Bk0 = B + (size_t)ka * N + col0 + m;
        const float* Bk1 = Bk0 + N;
        #pragma unroll
        for (int ct = 0; ct < 4; ++ct) {
            v2f b;
            b.x = Bk0[ct * 16];
            b.y = Bk1[ct * 16];
            acc[0][ct] = __builtin_amdgcn_wmma_f32_16x16x4_f32(false, a0, false, b,
                                                               (short)0, acc[0][ct],
                                                               false, false);
            acc[1][ct] = __builtin_amdgcn_wmma_f32_16x16x4_f32(false, a1, false, b,
                                                               (short)0, acc[1][ct],
                                                               false, false);
        }
    }

    #pragma unroll
    for (int rt = 0; rt < 2; ++rt) {
        #pragma unroll
        for (int ct = 0; ct < 4; ++ct) {
            const int cc = col0 + ct * 16 + m;
            const float bv = (mode >= 1) ? bias[cc] : 0.0f;
            #pragma unroll
            for (int i = 0; i < 8; ++i) {
                const int rr = row0 + rt * 16 + i + 8 * h;  // C layout: M = i+8h, N = m
                float v = acc[rt][ct][i] + bv;
                if (mode == 2) v = gelu_exact(v);
                C[(size_t)rr * N + cc] = v;
            }
        }
    }
}

// -------------------------------------------------------------------------
// block(128) tree reduction helper
// -------------------------------------------------------------------------
__device__ __forceinline__ float block_reduce_sum128(float v, float* sbuf)
{
    const int t = threadIdx.x;
    sbuf[t] = v;
    __syncthreads();
    #pragma unroll
    for (int s = 64; s > 0; s >>= 1) {
        if (t < s) sbuf[t] += sbuf[t + s];
        __syncthreads();
    }
    float r = sbuf[0];
    __syncthreads();
    return r;
}

// -------------------------------------------------------------------------
// Row-wise epilogue: l2norm(k), l2norm(q), alpha transforms, importance,
// path-weight softmax.   grid=ROWS, block=128
// -------------------------------------------------------------------------
__global__ __launch_bounds__(128)
void epilogue_kernel(const float* __restrict__ x,
                     float* __restrict__ Km, float* __restrict__ Qm,
                     float* __restrict__ AF, float* __restrict__ AS,
                     const float* __restrict__ IRh,
                     const float* __restrict__ fast_lambda,
                     const float* __restrict__ slow_lambda,
                     const float* __restrict__ ir_w2, const float* __restrict__ ir_b2,
                     const float* __restrict__ pw_w, const float* __restrict__ pw_b,
                     float* __restrict__ IMP, float* __restrict__ PW)
{
    __shared__ float sbuf[128];
    const int row = blockIdx.x;
    const int c   = threadIdx.x;
    const size_t base = (size_t)row * D_STATE + c;

    float kv = Km[base];
    float ks = block_reduce_sum128(kv * kv, sbuf);
    Km[base] = kv / fmaxf(sqrtf(ks), 1e-12f);

    float qv = Qm[base];
    float qs = block_reduce_sum128(qv * qv, sbuf);
    Qm[base] = qv / fmaxf(sqrtf(qs), 1e-12f);

    float laf = logf(sigmoidf_(fast_lambda[c]) + 1e-8f);
    float las = logf(sigmoidf_(slow_lambda[c]) + 1e-8f);
    AF[base] = expf(8.0f * sigmoidf_(AF[base]) * laf);
    AS[base] = expf(8.0f * sigmoidf_(AS[base]) * las);

    float isum = block_reduce_sum128(IRh[base] * ir_w2[c], sbuf);
    if (c == 0) IMP[row] = sigmoidf_(isum + ir_b2[0]);

    const float* xr = x + (size_t)row * D_MODEL;
    float p0 = 0.f, p1 = 0.f, p2 = 0.f;
    #pragma unroll
    for (int j = 0; j < 16; ++j) {
        const int idx = c * 16 + j;
        const float xv = xr[idx];
        p0 += xv * pw_w[idx * 3 + 0];
        p1 += xv * pw_w[idx * 3 + 1];
        p2 += xv * pw_w[idx * 3 + 2];
    }
    p0 = block_reduce_sum128(p0, sbuf);
    p1 = block_reduce_sum128(p1, sbuf);
    p2 = block_reduce_sum128(p2, sbuf);
    if (c == 0) {
        float l0 = p0 + pw_b[0], l1 = p1 + pw_b[1], l2 = p2 + pw_b[2];
        float mx = fmaxf(l0, fmaxf(l1, l2));
        float e0 = expf(l0 - mx), e1 = expf(l1 - mx), e2 = expf(l2 - mx);
        float inv = 1.0f / (e0 + e1 + e2);
        PW[(size_t)row * 3 + 0] = e0 * inv;
        PW[(size_t)row * 3 + 1] = e1 * inv;
        PW[(size_t)row * 3 + 2] = e2 * inv;
    }
}

// -------------------------------------------------------------------------
// Sequential scan. One 1024-thread workgroup (32 waves) per batch element.
// Thread (r=tid>>7, c=tid&127) owns rows [16r,16r+16) of column c of fH and
// sH in registers. Per-step vectors are double-buffered in LDS and filled
// by GLOBAL_LOAD_ASYNC_TO_LDS_B128 (ASYNCcnt) one step ahead when available.
// -------------------------------------------------------------------------
__global__ __launch_bounds__(1024)
void scan_kernel(const float* __restrict__ Km, const float* __restrict__ Vm,
                 const float* __restrict__ Qm, const float* __restrict__ AF,
                 const float* __restrict__ AS, const float* __restrict__ NU,
                 const float* __restrict__ IMP, const float* __restrict__ PW,
                 float* __restrict__ OUTR)
{
    const int b   = blockIdx.x;
    const int tid = threadIdx.x;
    const int c   = tid & 127;
    const int r   = tid >> 7;

    __shared__ float svec[2][6][D_STATE];   // double-buffered k,v,q,af,as,nu
    __shared__ float part[4][8][D_STATE];
    __shared__ float redv[4][D_STATE];      // pred_fast, pred_slow, kpf, kps
    __shared__ float snS[D_STATE];
    __shared__ float sscal[8];              // 0:sur 1:updn 2:upds 3..5:pw 6:imp
    __shared__ float werr[4];

    float fH[16], sH[16];
    #pragma unroll
    for (int j = 0; j < 16; ++j) { fH[j] = 0.f; sH[j] = 0.f; }
    if (r == 0) snS[c] = 0.f;
    __syncthreads();

#if HAVE_ASYNC_LDS
    // async issue helper inlined: threads 0..191 (6 waves) each move 16B
    #define ISSUE_ASYNC(step, buf_)                                            \
        if (tid < 192) {                                                       \
            const int a_ = tid >> 5;          /* wave-uniform array id */      \
            const int c4_ = (tid & 31) * 4;                                    \
            const float* gp_;                                                  \
            switch (a_) {                                                      \
                case 0:  gp_ = Km; break;                                      \
                case 1:  gp_ = Vm; break;                                      \
                case 2:  gp_ = Qm; break;                                      \
                case 3:  gp_ = AF; break;                                      \
                case 4:  gp_ = AS; break;                                      \
                default: gp_ = NU; break;                                      \
            }                                                                  \
            async_copy_b128(gp_ + ((size_t)b * SEQ + (step)) * D_STATE + c4_,  \
                            &svec[buf_][a_][c4_]);                             \
        }
    ISSUE_ASYNC(0, 0)
#endif

    for (int t = 0; t < SEQ; ++t) {
        const size_t row = (size_t)b * SEQ + t;
        const int buf = t & 1;

#if HAVE_ASYNC_LDS
        if (t + 1 < SEQ) { ISSUE_ASYNC(t + 1, buf ^ 1) }
#else
        if (tid < 768) {
            const int a = tid >> 7;     // uniform within each wave32
            const float* p;
            switch (a) {
                case 0:  p = Km; break;
                case 1:  p = Vm; break;
                case 2:  p = Qm; break;
                case 3:  p = AF; break;
                case 4:  p = AS; break;
                default: p = NU; break;
            }
            svec[buf][a][c] = p[row * D_STATE + c];
        }
#endif
        if (tid >= 768 && tid < 771) {
            sscal[3 + (tid - 768)] = PW[row * 3 + (tid - 768)];
        } else if (tid == 771) {
            sscal[6] = IMP[row];
        }
#if HAVE_ASYNC_LDS
        // current buffer's loads are the oldest outstanding; allow the newly
        // issued (t+1) batch to remain in flight.
        if (t + 1 < SEQ) { asm volatile("s_wait_asynccnt 0x1" ::: "memory"); }
        else             { asm volatile("s_wait_asynccnt 0x0" ::: "memory"); }
#endif
        __syncthreads();

        const float* kvec  = svec[buf][0];
        const float* vvec  = svec[buf][1];
        const float* qvec  = svec[buf][2];
        const float* afvec = svec[buf][3];
        const float* asvec = svec[buf][4];
        const float* nuvec = svec[buf][5];

        // ---- pass 1: pred_fast/slow and k'H from OLD state ----
        float pf = 0.f, ps = 0.f, gf = 0.f, gs = 0.f;
        #pragma unroll
        for (int j = 0; j < 16; ++j) {
            const int q = r * 16 + j;
            const float qq = qvec[q], kk = kvec[q];
            pf += qq * fH[j];  ps += qq * sH[j];
            gf += kk * fH[j];  gs += kk * sH[j];
        }
        part[0][r][c] = pf; part[1][r][c] = ps;
        part[2][r][c] = gf; part[3][r][c] = gs;
        __syncthreads();

        if (tid < 512) {
            const int a = tid >> 7;
            float s = 0.f;
            #pragma unroll
            for (int rr = 0; rr < 8; ++rr) s += part[a][rr][c];
            redv[a][c] = s;
        }
        __syncthreads();

        // ---- surprise: err = sum_v (v - 0.5*(pf+ps))^2 ----
        if (tid < 128) {
            float d = vvec[c] - 0.5f * (redv[0][c] + redv[1][c]);
            float e = d * d;
            #pragma unroll
            for (int off = 16; off > 0; off >>= 1) e += __shfl_down(e, off, 32);
            if ((tid & 31) == 0) werr[tid >> 5] = e;
        }
        __syncthreads();
        if (tid == 0) {
            const float err = werr[0] + werr[1] + werr[2] + werr[3];
            const float sur = 1.0f / (1.0f + expf(-err / (1.0f + 1e-6f)));
            sscal[0] = sur;
            sscal[1] = (sur > 0.7f)      ? 1.0f : 0.0f;
            sscal[2] = (sscal[6] > 0.5f) ? 1.0f : 0.0f;
        }
        __syncthreads();

        // ---- state update + output contraction against NEW state ----
        const float sur  = sscal[0];
        const float updn = sscal[1];
        const float upds = sscal[2];
        const float vtc  = vvec[c];
        const float dfc  = vtc - redv[2][c];
        const float dsc  = vtc - redv[3][c];

        float of = 0.f, os = 0.f;
        #pragma unroll
        for (int j = 0; j < 16; ++j) {
            const int q  = r * 16 + j;
            const float af  = afvec[q];
            const float as_ = asvec[q];
            const float kq  = kvec[q];
            const float qq  = qvec[q];
            fH[j] = af  * fH[j] + (1.0f - af)  * (sur * kq * dfc);
            sH[j] = as_ * sH[j] + (1.0f - as_) * (upds * sur * kq * dsc);
            of += qq * fH[j];
            os += qq * sH[j];
        }
        part[0][r][c] = of;
        part[1][r][c] = os;
        if (r == 0) {
            float n = snS[c];
            n = n + updn * (nuvec[c] - n) * 0.1f;
            snS[c] = n;
        }
        __syncthreads();

        if (tid < 256) {
            const int a = tid >> 7;
            float s = 0.f;
            #pragma unroll
            for (int rr = 0; rr < 8; ++rr) s += part[a][rr][c];
            redv[a][c] = s;
        }
        __syncthreads();

        if (tid < 128) {
            const float out = sscal[3] * redv[0][c] + sscal[4] * redv[1][c]
                            + sscal[5] * snS[c];
            OUTR[row * D_STATE + c] = out;
        }
        __syncthreads();
    }
}

// -------------------------------------------------------------------------
// RMS norm over last dim (128) * norm_w.  grid=ROWS, block=128
// -------------------------------------------------------------------------
__global__ __launch_bounds__(128)
void rms_kernel(const float* __restrict__ OUTR, const float* __restrict__ norm_w,
                float* __restrict__ NRM)
{
    __shared__ float sbuf[128];
    const int row = blockIdx.x;
    const int c   = threadIdx.x;
    const float o = OUTR[(size_t)row * D_STATE + c];
    const float ss = block_reduce_sum128(o * o, sbuf);
    const float rr = rsqrtf(ss * (1.0f / (float)D_STATE) + 1e-6f);
    NRM[(size_t)row * D_STATE + c] = o * rr * norm_w[c];
}

// -------------------------------------------------------------------------
// launch
// -------------------------------------------------------------------------
extern "C" void kernel_launch(void* const* d_in, const int* in_sizes, int n_in,
                              void* d_out, int out_size, void* d_ws, size_t ws_size,
                              hipStream_t stream)
{
    (void)in_sizes; (void)n_in; (void)out_size; (void)ws_size;

    const float* x           = (const float*)d_in[0];
    const float* Wk          = (const float*)d_in[1];
    const float* Wv          = (const float*)d_in[2];
    const float* Wq          = (const float*)d_in[3];
    const float* fast_lambda = (const float*)d_in[4];
    const float* slow_lambda = (const float*)d_in[5];
    const float* Waf         = (const float*)d_in[6];
    const float* baf         = (const float*)d_in[7];
    const float* Was         = (const float*)d_in[8];
    const float* bas         = (const float*)d_in[9];
    const float* nm_w1       = (const float*)d_in[10];
    const float* nm_b1       = (const float*)d_in[11];
    const float* nm_w2       = (const float*)d_in[12];
    const float* nm_b2       = (const float*)d_in[13];
    const float* ir_w1       = (const float*)d_in[14];
    const float* ir_b1       = (const float*)d_in[15];
    const float* ir_w2       = (const float*)d_in[16];
    const float* ir_b2       = (const float*)d_in[17];
    const float* pw_w        = (const float*)d_in[18];
    const float* pw_b        = (const float*)d_in[19];
    const float* Wo          = (const float*)d_in[20];
    const float* norm_w      = (const float*)d_in[21];
    float* out = (float*)d_out;

    // workspace layout (floats)
    float* ws = (float*)d_ws;
    const size_t RD = (size_t)ROWS * D_STATE;       // 8192*128
    float* Km   = ws;             ws += RD;
    float* Vm   = ws;             ws += RD;
    float* Qm   = ws;             ws += RD;
    float* AF   = ws;             ws += RD;
    float* AS   = ws;             ws += RD;
    float* IRh  = ws;             ws += RD;
    float* H1   = ws;             ws += (size_t)ROWS * 256;
    float* NU   = ws;             ws += RD;
    float* IMP  = ws;             ws += ROWS;
    float* PW   = ws;             ws += (size_t)ROWS * 3;
    float* OUTR = ws;             ws += RD;
    float* NRM  = ws;             ws += RD;

    const dim3 gblk(32);   // one wave per workgroup for GEMM

    // Phase A: projections (fp32 WMMA GEMMs, 32x64 per wave)
    gemm_wmma_f32_kernel<<<dim3(ROWS/32, D_STATE/64), gblk, 0, stream>>>(x, Wk, nullptr, Km, ROWS, D_STATE, D_MODEL, 0);
    gemm_wmma_f32_kernel<<<dim3(ROWS/32, D_STATE/64), gblk, 0, stream>>>(x, Wv, nullptr, Vm, ROWS, D_STATE, D_MODEL, 0);
    gemm_wmma_f32_kernel<<<dim3(ROWS/32, D_STATE/64), gblk, 0, stream>>>(x, Wq, nullptr, Qm, ROWS, D_STATE, D_MODEL, 0);
    gemm_wmma_f32_kernel<<<dim3(ROWS/32, D_STATE/64), gblk, 0, stream>>>(x, Waf, baf, AF, ROWS, D_STATE, D_MODEL, 1);
    gemm_wmma_f32_kernel<<<dim3(ROWS/32, D_STATE/64), gblk, 0, stream>>>(x, Was, bas, AS, ROWS, D_STATE, D_MODEL, 1);
    gemm_wmma_f32_kernel<<<dim3(ROWS/32, D_STATE/64), gblk, 0, stream>>>(x, ir_w1, ir_b1, IRh, ROWS, D_STATE, D_MODEL, 2);

    // neural memory MLP: gelu(V @ nm_w1 + b1) @ nm_w2 + b2
    gemm_wmma_f32_kernel<<<dim3(ROWS/32, 256/64), gblk, 0, stream>>>(Vm, nm_w1, nm_b1, H1, ROWS, 256, D_STATE, 2);
    gemm_wmma_f32_kernel<<<dim3(ROWS/32, D_STATE/64), gblk, 0, stream>>>(H1, nm_w2, nm_b2, NU, ROWS, D_STATE, 256, 1);

    // row-wise epilogues
    epilogue_kernel<<<ROWS, 128, 0, stream>>>(x, Km, Qm, AF, AS, IRh,
                                              fast_lambda, slow_lambda,
                                              ir_w2, ir_b2, pw_w, pw_b, IMP, PW);

    // Phase B: sequential scan, one WGP-resident workgroup per batch element
    scan_kernel<<<BATCH, 1024, 0, stream>>>(Km, Vm, Qm, AF, AS, NU, IMP, PW, OUTR);

    // Phase C: RMS norm + output projection
    rms_kernel<<<ROWS, 128, 0, stream>>>(OUTR, norm_w, NRM);
    gemm_wmma_f32_kernel<<<dim3(ROWS/32, D_MODEL/64), gblk, 0, stream>>>(NRM, Wo, nullptr, out, ROWS, D_MODEL, D_STATE, 0);
}